// SinkhornOT_19550691131568
// MI455X (gfx1250) — compile-verified
//
#include <hip/hip_runtime.h>
#include <hip/hip_bf16.h>
#include <math.h>

// ---------------------------------------------------------------------------
// Sinkhorn OT pipeline for MI455X (gfx1250, wave32, WMMA f32 16x16x4).
// B=256, K=M=64, D=256.
// ---------------------------------------------------------------------------

typedef float v2f __attribute__((ext_vector_type(2)));
typedef float v8f __attribute__((ext_vector_type(8)));

#define DIMD   256
#define NB     256
#define NK     64
#define NM     64
#define NROWS  32768            // (B*K + B*M) projected rows
#define EPS    0.05f
#define XP     260              // LDS pitch (floats) to dodge 64-bank conflicts

static __device__ __forceinline__ v8f wmma_f32(v2f a, v2f b, v8f c) {
  // D = A(16x4,f32) * B(4x16,f32) + C(16x16,f32)   -> v_wmma_f32_16x16x4_f32
  return __builtin_amdgcn_wmma_f32_16x16x4_f32(false, a, false, b,
                                               (short)0, c, false, false);
}

// ---------------------------------------------------------------------------
// Kernel 1: fused projection  P = relu(X@W1 + b1)@W2 + b2
// grid = NROWS/16 blocks, block = 512 threads (16 waves).
// Block handles a 16-row strip; wave w owns output columns [16w,16w+16).
// ---------------------------------------------------------------------------
__global__ __launch_bounds__(512) void proj_kernel(
    const float* __restrict__ q, const float* __restrict__ r,
    const float* __restrict__ W1, const float* __restrict__ b1,
    const float* __restrict__ W2, const float* __restrict__ b2,
    float* __restrict__ P) {
  __shared__ float Xs[16 * XP];
  __shared__ float Hs[16 * XP];

  const int tid  = threadIdx.x;
  const int blk  = blockIdx.x;
  const int row0 = blk * 16;
  const float* X = (row0 < 16384) ? (q + (size_t)row0 * DIMD)
                                  : (r + (size_t)(row0 - 16384) * DIMD);

  // Stage 16x256 fp32 tile into LDS with padded pitch (float4 copies).
  for (int i = tid; i < 16 * (DIMD / 4); i += 512) {
    const int rr = i >> 6;        // row 0..15
    const int cc = i & 63;        // float4 column
    ((float4*)(Xs + rr * XP))[cc] = ((const float4*)(X + rr * DIMD))[cc];
  }
  __syncthreads();

  const int wave  = tid >> 5;
  const int lane  = tid & 31;
  const int nbase = wave * 16;
  const int ml    = lane & 15;
  const int koff  = (lane < 16) ? 0 : 2;
  const int mrow  = (lane < 16) ? 0 : 8;   // C/D layout: lanes 16-31 hold M=v+8

  // GEMM1: H = relu(X @ W1 + b1)
  v8f acc = {};
  for (int k0 = 0; k0 < DIMD; k0 += 4) {
    v2f a, b;
    a.x = Xs[ml * XP + k0 + koff];
    a.y = Xs[ml * XP + k0 + koff + 1];
    b.x = W1[(k0 + koff) * DIMD + nbase + ml];
    b.y = W1[(k0 + koff + 1) * DIMD + nbase + ml];
    acc = wmma_f32(a, b, acc);
  }
  const float bias1 = b1[nbase + ml];
#pragma unroll
  for (int v = 0; v < 8; ++v) {
    float h = acc[v] + bias1;
    Hs[(mrow + v) * XP + nbase + ml] = h > 0.f ? h : 0.f;
  }
  __syncthreads();

  // GEMM2: Pstrip = H @ W2 + b2
  v8f acc2 = {};
  for (int k0 = 0; k0 < DIMD; k0 += 4) {
    v2f a, b;
    a.x = Hs[ml * XP + k0 + koff];
    a.y = Hs[ml * XP + k0 + koff + 1];
    b.x = W2[(k0 + koff) * DIMD + nbase + ml];
    b.y = W2[(k0 + koff + 1) * DIMD + nbase + ml];
    acc2 = wmma_f32(a, b, acc2);
  }
  const float bias2 = b2[nbase + ml];
  float* Pout = P + (size_t)row0 * DIMD;
#pragma unroll
  for (int v = 0; v < 8; ++v)
    Pout[(mrow + v) * DIMD + nbase + ml] = acc2[v] + bias2;
}

// ---------------------------------------------------------------------------
// Kernel 2: row squared-norms of P (32768 rows of 256).
// block = 256 (8 waves), wave per row, grid = 4096.
// ---------------------------------------------------------------------------
__global__ __launch_bounds__(256) void norm_kernel(const float* __restrict__ P,
                                                   float* __restrict__ norms) {
  const int row  = blockIdx.x * 8 + (threadIdx.x >> 5);
  const int lane = threadIdx.x & 31;
  const float* p = P + (size_t)row * DIMD;
  float s = 0.f;
#pragma unroll
  for (int i = lane; i < DIMD; i += 32) { float v = p[i]; s += v * v; }
#pragma unroll
  for (int off = 16; off; off >>= 1) s += __shfl_xor(s, off, 32);
  if (lane == 0) norms[row] = s;
}

// ---------------------------------------------------------------------------
// Kernel 3: cost matrix.  Per batch: G = pq[b] @ pr[b]^T via WMMA, then
// C = sqrt(max(nq + nr - 2G, 1e-6)).  grid = 256, block = 512 (16 waves,
// wave w -> 16x16 tile (w/4, w%4) of the 64x64 output).
// ---------------------------------------------------------------------------
__global__ __launch_bounds__(512) void cost_kernel(
    const float* __restrict__ P, const float* __restrict__ norms,
    float* __restrict__ Cout) {
  const int b = blockIdx.x;
  const float* pq = P + (size_t)b * NK * DIMD;
  const float* pr = P + (size_t)(16384 + b * NM) * DIMD;

  const int tid  = threadIdx.x;
  const int wave = tid >> 5;
  const int lane = tid & 31;
  const int ti   = wave >> 2;          // output row tile
  const int tj   = wave & 3;           // output col tile
  const int ml   = lane & 15;
  const int koff = (lane < 16) ? 0 : 2;
  const int mrow = (lane < 16) ? 0 : 8;

  v8f acc = {};
  for (int k0 = 0; k0 < DIMD; k0 += 4) {
    v2f a, bfr;
    a.x   = pq[(ti * 16 + ml) * DIMD + k0 + koff];
    a.y   = pq[(ti * 16 + ml) * DIMD + k0 + koff + 1];
    // B(k,n) = pr^T(k,n) = pr[n][k]
    bfr.x = pr[(tj * 16 + ml) * DIMD + k0 + koff];
    bfr.y = pr[(tj * 16 + ml) * DIMD + k0 + koff + 1];
    acc = wmma_f32(a, bfr, acc);
  }
  const float nr_n = norms[16384 + b * NM + tj * 16 + ml];
#pragma unroll
  for (int v = 0; v < 8; ++v) {
    const int mm = ti * 16 + mrow + v;
    float d2 = norms[b * NK + mm] + nr_n - 2.f * acc[v];
    d2 = fmaxf(d2, 1e-6f);
    Cout[b * (NK * NM) + mm * NM + tj * 16 + ml] = sqrtf(d2);
  }
}

// ---------------------------------------------------------------------------
// Kernel 4: normalized log masses.  grid = 512 (256 mu + 256 nu), block = 32.
// ---------------------------------------------------------------------------
__global__ __launch_bounds__(32) void logmass_kernel(
    const float* __restrict__ mq, const float* __restrict__ mr,
    float* __restrict__ logmu, float* __restrict__ lognu) {
  const int idx  = blockIdx.x;
  const int lane = threadIdx.x;
  const float* src;
  float* dst;
  if (idx < NB) { src = mq + idx * NK;        dst = logmu + idx * NK; }
  else          { src = mr + (idx - NB) * NM; dst = lognu + (idx - NB) * NM; }

  float l0 = logf(fmaxf(src[lane],      1e-8f));
  float l1 = logf(fmaxf(src[lane + 32], 1e-8f));
  float mx = fmaxf(l0, l1);
#pragma unroll
  for (int off = 16; off; off >>= 1) mx = fmaxf(mx, __shfl_xor(mx, off, 32));
  float s = expf(l0 - mx) + expf(l1 - mx);
#pragma unroll
  for (int off = 16; off; off >>= 1) s += __shfl_xor(s, off, 32);
  const float lse = mx + logf(s);
  dst[lane]      = l0 - lse;
  dst[lane + 32] = l1 - lse;
}

// ---------------------------------------------------------------------------
// Kernel 5: Sinkhorn iterations + mesh refinement + cost/similarity.
// grid = 256 (one block per batch), block = 64 (2 waves); 64x64 matrices in
// LDS with pitch 65 so row and column walks are both bank-conflict-free.
// ---------------------------------------------------------------------------
#define LKP 65
__global__ __launch_bounds__(64) void sinkhorn_kernel(
    const float* __restrict__ Cmat, const float* __restrict__ logmu,
    const float* __restrict__ lognu, float* __restrict__ Tout,
    float* __restrict__ sim, float* __restrict__ costOut) {
  __shared__ float lk[NK * LKP];
  __shared__ float Tsh[NK * LKP];
  __shared__ float la_sh[NK];
  __shared__ float lb_sh[NM];
  __shared__ float red[NK];

  const int b = blockIdx.x;
  const int t = threadIdx.x;                 // 0..63
  const float inv_eps = 1.0f / EPS;
  const float* Cb = Cmat + (size_t)b * (NK * NM);

  const float lmu_t = logmu[b * NK + t];
  for (int m = 0; m < NM; ++m)
    lk[t * LKP + m] = -Cb[t * NM + m] * inv_eps + lmu_t + lognu[b * NM + m];
  lb_sh[t] = 0.f;
  __syncthreads();

  float la = 0.f;
  for (int it = 0; it < 15; ++it) {
    // log_a[t] = -logsumexp_m( lk[t][m] + log_b[m] )
    float mx = -3.4e38f;
    for (int m = 0; m < NM; ++m)
      mx = fmaxf(mx, lk[t * LKP + m] + lb_sh[m]);
    float s = 0.f;
    for (int m = 0; m < NM; ++m)
      s += expf(lk[t * LKP + m] + lb_sh[m] - mx);
    la = -(mx + logf(s));
    la_sh[t] = la;
    __syncthreads();
    // log_b[t] = -logsumexp_k( lk[k][t] + log_a[k] )
    mx = -3.4e38f;
    for (int k = 0; k < NK; ++k)
      mx = fmaxf(mx, lk[k * LKP + t] + la_sh[k]);
    s = 0.f;
    for (int k = 0; k < NK; ++k)
      s += expf(lk[k * LKP + t] + la_sh[k] - mx);
    lb_sh[t] = -(mx + logf(s));
    __syncthreads();
  }

  // T = exp(logK + log_a + log_b)
  for (int m = 0; m < NM; ++m)
    Tsh[t * LKP + m] = expf(lk[t * LKP + m] + la + lb_sh[m]);
  __syncthreads();

  // Mesh refinement: 3x (square, row-normalize, col-normalize)
  for (int it = 0; it < 3; ++it) {
    float rs = 0.f;
    for (int m = 0; m < NM; ++m) {
      float v = Tsh[t * LKP + m];
      v = v * v;
      Tsh[t * LKP + m] = v;
      rs += v;
    }
    const float rinv = 1.f / (rs + 1e-8f);
    for (int m = 0; m < NM; ++m) Tsh[t * LKP + m] *= rinv;
    __syncthreads();
    float cs = 0.f;
    for (int k = 0; k < NK; ++k) cs += Tsh[k * LKP + t];
    const float cinv = 1.f / (cs + 1e-8f);
    for (int k = 0; k < NK; ++k) Tsh[k * LKP + t] *= cinv;
    __syncthreads();
  }

  // cost = sum(T*C); write T out.
  float part = 0.f;
  for (int m = 0; m < NM; ++m) {
    const float tv = Tsh[t * LKP + m];
    part += tv * Cb[t * NM + m];
    Tout[b * (NK * NM) + t * NM + m] = tv;
  }
  red[t] = part;
  __syncthreads();
  if (t < 32) red[t] += red[t + 32];
  __syncthreads();
  if (t < 32) {
    float s = red[t];
#pragma unroll
    for (int off = 16; off; off >>= 1) s += __shfl_xor(s, off, 32);
    if (t == 0) {
      costOut[b] = s;
      sim[b] = 1.f / (1.f + expf(s));   // sigmoid(-cost)
    }
  }
}

// ---------------------------------------------------------------------------
extern "C" void kernel_launch(void* const* d_in, const int* in_sizes, int n_in,
                              void* d_out, int out_size, void* d_ws, size_t ws_size,
                              hipStream_t stream) {
  const float* slots_q = (const float*)d_in[0];
  const float* slots_r = (const float*)d_in[1];
  const float* mask_q  = (const float*)d_in[2];
  const float* mask_r  = (const float*)d_in[3];
  const float* W1      = (const float*)d_in[4];
  const float* b1      = (const float*)d_in[5];
  const float* W2      = (const float*)d_in[6];
  const float* b2      = (const float*)d_in[7];

  float* out      = (float*)d_out;
  float* out_sim  = out;                       // [256]
  float* out_T    = out + NB;                  // [256*64*64]
  float* out_C    = out_T + NB * NK * NM;      // [256*64*64]
  float* out_cost = out_C + NB * NK * NM;      // [256]

  float* ws    = (float*)d_ws;
  float* P     = ws;                           // 32768*256
  float* norms = P + (size_t)NROWS * DIMD;     // 32768
  float* logmu = norms + NROWS;                // 16384
  float* lognu = logmu + NB * NK;              // 16384

  proj_kernel<<<NROWS / 16, 512, 0, stream>>>(slots_q, slots_r, W1, b1, W2, b2, P);
  norm_kernel<<<NROWS / 8, 256, 0, stream>>>(P, norms);
  cost_kernel<<<NB, 512, 0, stream>>>(P, norms, out_C);
  logmass_kernel<<<2 * NB, 32, 0, stream>>>(mask_q, mask_r, logmu, lognu);
  sinkhorn_kernel<<<NB, 64, 0, stream>>>(out_C, logmu, lognu, out_T, out_sim, out_cost);
}